// GroupOfGESNCell_81037442941025
// MI455X (gfx1250) — compile-verified
//
#include <hip/hip_runtime.h>
#include <math.h>

typedef __attribute__((ext_vector_type(2))) float v2f;
typedef __attribute__((ext_vector_type(8))) float v8f;

#define NN 8192
#define FF 128
#define HH 64
#define GG 4
#define GH 256          // G*H fused column dimension
#define BM 64
#define BK 64
#define LDA 68          // As row stride (words): conflict-free, 8B-aligned b64 reads
#define LDB2 160        // Bs2 k-pair row stride (128 data + 32 pad => +32 mod 64)
#define NTHR 128        // 4 waves, each computes a 32x32 output block

// workspace layout (floats)
#define WS_LIN  ((long)NN * GH)
#define WS_CTRL (2L * NN * GH)   // [0..3] diffsq (float), [4..7] done (int)

// ---------------------------------------------------------------------------
// Synchronous tile movers (used by the small GEMMs): 64x64 fp32, 128 threads
// ---------------------------------------------------------------------------
__device__ __forceinline__ void pref_tile(float4* p, const float* src, long ld) {
  const int t = threadIdx.x;
#pragma unroll
  for (int i = 0; i < 8; ++i) {
    const int s  = t + i * NTHR;
    const int r  = s >> 4;
    const int cq = (s & 15) << 2;
    p[i] = *(const float4*)(src + (long)r * ld + cq);
  }
}

__device__ __forceinline__ void st_tileA(float* As, const float4* p) {
  const int t = threadIdx.x;
#pragma unroll
  for (int i = 0; i < 8; ++i) {
    const int s  = t + i * NTHR;
    const int r  = s >> 4;
    const int cq = (s & 15) << 2;
    *(float4*)(As + r * LDA + cq) = p[i];   // (r*68+cq)*4 is 16B aligned
  }
}

// Transposed B fill: Bs2 pair(k)/col(c) from src[c*ld + k] (contiguous in k)
__device__ __forceinline__ void ld_tileBt(float* Bs2, const float* src, long ld) {
  const int t = threadIdx.x;
#pragma unroll
  for (int i = 0; i < 8; ++i) {
    const int s  = t + i * NTHR;
    const int c  = s >> 4;             // dst column 0..63
    const int kq = (s & 15) << 2;      // k 0..60 step 4 (even)
    const float4 v = *(const float4*)(src + (long)c * ld + kq);
    float* d = Bs2 + (kq >> 1) * LDB2 + (c << 1);
    d[0] = v.x; d[1] = v.y;            // pair row kq/2
    d[LDB2] = v.z; d[LDB2 + 1] = v.w;  // pair row kq/2 + 1
  }
}

// ---------------------------------------------------------------------------
// Async staging for the big GEMM: global -> LDS without touching VGPR data.
// A: 64 rows x 256B contiguous.  B: 32 pair-rows x 512B contiguous (glin is
// stored K-paired in global memory, so pieces map 1:1 into the LDS layout).
// ---------------------------------------------------------------------------
__device__ __forceinline__ void async_ld16(unsigned la, unsigned long long ga) {
  asm volatile("global_load_async_to_lds_b128 %0, %1, off"
               :: "v"(la), "v"(ga) : "memory");
}

__device__ __forceinline__ void async_chunk(const float* Lrow, const float* glin,
                                            int k0, int col0,
                                            unsigned ldsA, unsigned ldsB) {
  const int t = threadIdx.x;
#pragma unroll
  for (int i = 0; i < 8; ++i) {                  // A tile: 1024 x 16B pieces
    const int s  = t + i * NTHR;
    const int r  = s >> 4;
    const int q4 = (s & 15) << 2;
    const unsigned long long ga =
        (unsigned long long)(size_t)(Lrow + (long)r * NN + k0 + q4);
    async_ld16(ldsA + (unsigned)((r * LDA + q4) << 2), ga);
  }
#pragma unroll
  for (int i = 0; i < 8; ++i) {                  // B tile: 1024 x 16B pieces
    const int s  = t + i * NTHR;
    const int p  = s >> 5;                       // pair-row 0..31
    const int q4 = (s & 31) << 2;                // word offset within pair-row
    const unsigned long long gb =
        (unsigned long long)(size_t)(glin + ((long)(k0 >> 1) + p) * (2 * GH)
                                     + (col0 << 1) + q4);
    async_ld16(ldsB + (unsigned)((p * LDB2 + q4) << 2), gb);
  }
}

// ---------------------------------------------------------------------------
// One K=64 chunk: each wave does 2x2 16x16 subtiles (4 WMMAs / 4 b64 loads
// per K-step).  As[m][k] stride LDA; Bs2 K-paired, stride LDB2.
// ---------------------------------------------------------------------------
__device__ __forceinline__ void mma4(const float* As, const float* Bs2,
                                     int mt, int nt, v8f acc[4]) {
  const int lane = threadIdx.x & 31;
  const int m    = lane & 15;          // M (for A) / N (for B)
  const int kl   = (lane >> 4) << 1;   // lane K offset: 0 or 2
  const float* ap = As + (mt * 32 + m) * LDA + kl;
  const float* bp = Bs2 + (kl >> 1) * LDB2 + ((nt * 32 + m) << 1);
#pragma unroll
  for (int kk = 0; kk < BK; kk += 4) {
    const v2f a0 = *(const v2f*)(ap + kk);
    const v2f a1 = *(const v2f*)(ap + kk + 16 * LDA);
    const v2f b0 = *(const v2f*)(bp + (kk >> 1) * LDB2);
    const v2f b1 = *(const v2f*)(bp + (kk >> 1) * LDB2 + 32);
    acc[0] = __builtin_amdgcn_wmma_f32_16x16x4_f32(false, a0, false, b0, (short)0, acc[0], false, false);
    acc[1] = __builtin_amdgcn_wmma_f32_16x16x4_f32(false, a0, false, b1, (short)0, acc[1], false, false);
    acc[2] = __builtin_amdgcn_wmma_f32_16x16x4_f32(false, a1, false, b0, (short)0, acc[2], false, false);
    acc[3] = __builtin_amdgcn_wmma_f32_16x16x4_f32(false, a1, false, b1, (short)0, acc[3], false, false);
  }
}

// ---------------------------------------------------------------------------
// init: zero hx (lives in d_out) and control block
// ---------------------------------------------------------------------------
__global__ void k_init(float* __restrict__ hx, float* __restrict__ diffsq, int* __restrict__ done) {
  const long t = (long)blockIdx.x * 256 + threadIdx.x;
  if (t < (long)NN * GH) hx[t] = 0.0f;
  if (t < GG) { diffsq[t] = 0.0f; done[t] = 0; }
}

// ---------------------------------------------------------------------------
// wiu[n][g*64+h] = sum_f X[n][f] * W_ih[g][h][f]     (K = 128, two chunks)
// ---------------------------------------------------------------------------
__global__ void k_wiu(const float* __restrict__ X, const float* __restrict__ Wih,
                      float* __restrict__ wiu) {
  __shared__ float As[BM * LDA];
  __shared__ float Bs2[(BK / 2) * LDB2];
  const int row0 = blockIdx.x * BM;
  const int g    = blockIdx.y;
  const int wv   = threadIdx.x >> 5;
  const int mt   = wv >> 1;
  const int nt   = wv & 1;

  v8f acc[4] = {};
  float4 pa[8];
  for (int k0 = 0; k0 < FF; k0 += BK) {
    pref_tile(pa, X + (long)row0 * FF + k0, FF);
    st_tileA(As, pa);
    ld_tileBt(Bs2, Wih + (long)g * HH * FF + k0, FF);   // B[f][h] = Wih[g][h][k0+f]
    __syncthreads();
    mma4(As, Bs2, mt, nt, acc);
    __syncthreads();
  }
  const int lane = threadIdx.x & 31;
  const int n    = lane & 15;
  const int rb   = row0 + mt * 32 + ((lane >> 4) << 3);
  const int cb   = g * HH + nt * 32 + n;
#pragma unroll
  for (int jr = 0; jr < 2; ++jr)
#pragma unroll
    for (int i = 0; i < 8; ++i) {
      const long r = rb + jr * 16 + i;
      wiu[r * GH + cb]      = acc[jr * 2 + 0][i];
      wiu[r * GH + cb + 16] = acc[jr * 2 + 1][i];
    }
}

// ---------------------------------------------------------------------------
// lin = hx @ W_hh[g]^T, written to global in K-PAIRED layout:
//   glin[(m>>1)*512 + 2*c + (m&1)] = lin[m][c]
// ---------------------------------------------------------------------------
__global__ void k_lin(const float* __restrict__ hx, const float* __restrict__ Whh,
                      float* __restrict__ glin) {
  __shared__ float As[BM * LDA];
  __shared__ float Bs2[(BK / 2) * LDB2];
  const int row0 = blockIdx.x * BM;
  const int g    = blockIdx.y;
  const int wv   = threadIdx.x >> 5;
  const int mt   = wv >> 1;
  const int nt   = wv & 1;

  v8f acc[4] = {};
  float4 pa[8];
  pref_tile(pa, hx + (long)row0 * GH + g * HH, GH);     // As[m][h]
  st_tileA(As, pa);
  ld_tileBt(Bs2, Whh + (long)g * HH * HH, HH);          // B[h][k] = Whh[g][k][h]
  __syncthreads();
  mma4(As, Bs2, mt, nt, acc);
  __syncthreads();

  const int lane = threadIdx.x & 31;
  const int n    = lane & 15;
  const int rb   = row0 + mt * 32 + ((lane >> 4) << 3);
  const int cb   = g * HH + nt * 32 + n;
#pragma unroll
  for (int jr = 0; jr < 2; ++jr)
#pragma unroll
    for (int i = 0; i < 8; ++i) {
      const long r = rb + jr * 16 + i;
      const long p = (r >> 1) * (2 * GH);
      const int  e = (int)(r & 1);
      glin[p + ((long)cb << 1) + e]        = acc[jr * 2 + 0][i];
      glin[p + ((long)(cb + 16) << 1) + e] = acc[jr * 2 + 1][i];
    }
}

// ---------------------------------------------------------------------------
// Big fused kernel, async double-buffered:  C = L @ lin ; new = tanh(wiu + C)
//   diffsq[g] += sum (new - hx)^2 ;  if (!done[g]) hx = new
// ---------------------------------------------------------------------------
__global__ void k_big(const float* __restrict__ L, const float* __restrict__ glin,
                      const float* __restrict__ wiu, float* __restrict__ hx,
                      float* __restrict__ diffsq, const int* __restrict__ done) {
  __shared__ float As[2][BM * LDA];
  __shared__ float Bs2[2][(BK / 2) * LDB2];
  __shared__ float red[NTHR];
  const int row0 = blockIdx.x * BM;
  const int g    = blockIdx.y;
  const int col0 = g * HH;
  const int wv   = threadIdx.x >> 5;
  const int mt   = wv >> 1;
  const int nt   = wv & 1;
  const float* Lrow = L + (long)row0 * NN;

  const unsigned lA0 = (unsigned)(size_t)&As[0][0];
  const unsigned lA1 = (unsigned)(size_t)&As[1][0];
  const unsigned lB0 = (unsigned)(size_t)&Bs2[0][0];
  const unsigned lB1 = (unsigned)(size_t)&Bs2[1][0];

  v8f acc[4] = {};
  async_chunk(Lrow, glin, 0, col0, lA0, lB0);      // prologue fill, buffer 0
  int buf = 0;
  for (int k0 = 0; k0 < NN; k0 += BK) {
    asm volatile("s_wait_asynccnt 0x0" ::: "memory");   // own pieces landed
    __syncthreads();                                    // everyone's landed,
                                                        // everyone's prev MMA done
    const int kn = k0 + BK;
    if (kn < NN)                                        // fill other buffer while
      async_chunk(Lrow, glin, kn, col0,                 // this one computes
                  buf ? lA0 : lA1, buf ? lB0 : lB1);
    mma4(&As[buf][0], &Bs2[buf][0], mt, nt, acc);
    buf ^= 1;
  }

  const int gdone = done[g];
  const int lane  = threadIdx.x & 31;
  const int n     = lane & 15;
  const int rb    = row0 + mt * 32 + ((lane >> 4) << 3);
  const int cb    = col0 + nt * 32 + n;
  float ldiff = 0.0f;
#pragma unroll
  for (int jr = 0; jr < 2; ++jr)
#pragma unroll
    for (int i = 0; i < 8; ++i) {
      const long idx0 = (long)(rb + jr * 16 + i) * GH + cb;
      const long idx1 = idx0 + 16;
      const float nv0 = tanhf(wiu[idx0] + acc[jr * 2 + 0][i]);
      const float nv1 = tanhf(wiu[idx1] + acc[jr * 2 + 1][i]);
      const float o0 = hx[idx0], o1 = hx[idx1];
      const float d0 = nv0 - o0, d1 = nv1 - o1;
      ldiff += d0 * d0 + d1 * d1;
      if (!gdone) { hx[idx0] = nv0; hx[idx1] = nv1; }
    }

  red[threadIdx.x] = ldiff;
  __syncthreads();
#pragma unroll
  for (int s = NTHR / 2; s > 0; s >>= 1) {
    if (threadIdx.x < s) red[threadIdx.x] += red[threadIdx.x + s];
    __syncthreads();
  }
  if (threadIdx.x == 0) atomicAdd(&diffsq[g], red[0]);
}

// ---------------------------------------------------------------------------
// done[g] |= (||diff||^2 < (1e-5)^2);  reset accumulator
// ---------------------------------------------------------------------------
__global__ void k_flag(float* __restrict__ diffsq, int* __restrict__ done) {
  const int t = threadIdx.x;
  if (t < GG) {
    if (!done[t] && diffsq[t] < 1e-10f) done[t] = 1;
    diffsq[t] = 0.0f;
  }
}

// ---------------------------------------------------------------------------
extern "C" void kernel_launch(void* const* d_in, const int* in_sizes, int n_in,
                              void* d_out, int out_size, void* d_ws, size_t ws_size,
                              hipStream_t stream) {
  const float* X   = (const float*)d_in[0];   // [N,F]
  const float* L   = (const float*)d_in[1];   // [N,N]
  const float* Wih = (const float*)d_in[2];   // [G,H,F]
  const float* Whh = (const float*)d_in[3];   // [G,H,H]

  float* hx  = (float*)d_out;                 // [N, G*H] output == state
  float* ws  = (float*)d_ws;
  float* wiu = ws;                            // [N, G*H] row-major
  float* lin = ws + WS_LIN;                   // [N, G*H] K-paired layout
  float* diffsq = ws + WS_CTRL;               // [G]
  int*   done   = (int*)(ws + WS_CTRL + GG);  // [G]

  const dim3 grid(NN / BM, GG);
  k_init<<<(NN * GH + 255) / 256, 256, 0, stream>>>(hx, diffsq, done);
  k_wiu<<<grid, NTHR, 0, stream>>>(X, Wih, wiu);
  for (int it = 0; it < 5; ++it) {
    k_lin<<<grid, NTHR, 0, stream>>>(hx, Whh, lin);
    k_big<<<grid, NTHR, 0, stream>>>(L, lin, wiu, hx, diffsq, done);
    k_flag<<<1, 32, 0, stream>>>(diffsq, done);
  }
}